// GCN_54271206752667
// MI455X (gfx1250) — compile-verified
//
#include <hip/hip_runtime.h>
#include <hip/hip_bf16.h>

typedef __attribute__((ext_vector_type(2))) float v2f;
typedef __attribute__((ext_vector_type(8))) float v8f;

#define NNODES 50000
#define NEDGES 800000
#define NFEAT  512
#define NHID   256
#define NCLASS 64

// ---------------------------------------------------------------------------
// GEMM1: t1[N,256] = dropout(x[N,512], mask1) @ W1[512,256]
// Block = 128 threads (4 wave32). Block computes a 16x256 output strip.
// A tile (16x512, dropout applied) staged in LDS with pad-516 rows.
// Each wave handles a 64-column quarter: 4 v8f accumulators,
// K-loop of 128 steps of V_WMMA_F32_16X16X4_F32.
// ---------------------------------------------------------------------------
__global__ void __launch_bounds__(128)
gcn_gemm1_dropout_wmma(const float* __restrict__ x,
                       const float* __restrict__ mask1,
                       const float* __restrict__ W1,
                       float* __restrict__ t1) {
    __shared__ float lds[16 * 516];
    const int tid = threadIdx.x;
    const int m0  = blockIdx.x * 16;

    // Cooperative coalesced load of 16x512 A tile with dropout (keep if u>=0.5, x2)
    for (int i = tid; i < 16 * 128; i += 128) {          // 128 float4 per row
        const int r = i >> 7;
        const int c = (i & 127) << 2;
        const size_t g = (size_t)(m0 + r) * NFEAT + c;
        const float4 xv = *(const float4*)(x + g);
        const float4 uv = *(const float4*)(mask1 + g);
        float* dst = lds + r * 516 + c;
        dst[0] = (uv.x >= 0.5f) ? xv.x * 2.0f : 0.0f;
        dst[1] = (uv.y >= 0.5f) ? xv.y * 2.0f : 0.0f;
        dst[2] = (uv.z >= 0.5f) ? xv.z * 2.0f : 0.0f;
        dst[3] = (uv.w >= 0.5f) ? xv.w * 2.0f : 0.0f;
    }
    __syncthreads();

    const int wave = tid >> 5;          // 0..3 -> 64-col quarter
    const int lane = tid & 31;
    const int half = lane >> 4;         // 0: K={0,1}, 1: K={2,3} (ISA 32-bit A/B layout)
    const int l16  = lane & 15;         // M (for A) / N (for B,C,D)
    const int n0   = wave * 64;

    v8f acc[4] = {v8f{}, v8f{}, v8f{}, v8f{}};

    for (int k = 0; k < NFEAT; k += 4) {
        const int kk = k + half * 2;
        const float* ap = lds + l16 * 516 + kk;
        v2f a;
        a.x = ap[0];
        a.y = ap[1];
#pragma unroll
        for (int j = 0; j < 4; ++j) {
            const int n = n0 + j * 16 + l16;
            v2f b;
            b.x = W1[(size_t)kk * NHID + n];
            b.y = W1[(size_t)(kk + 1) * NHID + n];
            acc[j] = __builtin_amdgcn_wmma_f32_16x16x4_f32(
                false, a, false, b, (short)0, acc[j], false, false);
        }
    }

    // C/D layout: VGPR v holds row M = v + half*8, col N = l16
#pragma unroll
    for (int j = 0; j < 4; ++j) {
        const int n = n0 + j * 16 + l16;
#pragma unroll
        for (int v = 0; v < 8; ++v) {
            t1[(size_t)(m0 + v + half * 8) * NHID + n] = acc[j][v];
        }
    }
}

// ---------------------------------------------------------------------------
// GEMM2: t2[N,64] = dropout(relu(h1[N,256]), mask2) @ W2[256,64]
// Block = 128 threads (4 waves), each wave one 16x16 tile; A tile pad-260.
// ---------------------------------------------------------------------------
__global__ void __launch_bounds__(128)
gcn_gemm2_relu_dropout_wmma(const float* __restrict__ h1,
                            const float* __restrict__ mask2,
                            const float* __restrict__ W2,
                            float* __restrict__ t2) {
    __shared__ float lds[16 * 260];
    const int tid = threadIdx.x;
    const int m0  = blockIdx.x * 16;

    for (int i = tid; i < 16 * 64; i += 128) {           // 64 float4 per row
        const int r = i >> 6;
        const int c = (i & 63) << 2;
        const size_t g = (size_t)(m0 + r) * NHID + c;
        const float4 hv = *(const float4*)(h1 + g);
        const float4 uv = *(const float4*)(mask2 + g);
        float* dst = lds + r * 260 + c;
        const float rx = hv.x > 0.0f ? hv.x : 0.0f;
        const float ry = hv.y > 0.0f ? hv.y : 0.0f;
        const float rz = hv.z > 0.0f ? hv.z : 0.0f;
        const float rw = hv.w > 0.0f ? hv.w : 0.0f;
        dst[0] = (uv.x >= 0.5f) ? rx * 2.0f : 0.0f;
        dst[1] = (uv.y >= 0.5f) ? ry * 2.0f : 0.0f;
        dst[2] = (uv.z >= 0.5f) ? rz * 2.0f : 0.0f;
        dst[3] = (uv.w >= 0.5f) ? rw * 2.0f : 0.0f;
    }
    __syncthreads();

    const int wave = tid >> 5;
    const int lane = tid & 31;
    const int half = lane >> 4;
    const int l16  = lane & 15;
    const int n    = wave * 16 + l16;   // N column within [0,64)

    v8f acc = {};

    for (int k = 0; k < NHID; k += 4) {
        const int kk = k + half * 2;
        const float* ap = lds + l16 * 260 + kk;
        v2f a;
        a.x = ap[0];
        a.y = ap[1];
        v2f b;
        b.x = W2[(size_t)kk * NCLASS + n];
        b.y = W2[(size_t)(kk + 1) * NCLASS + n];
        acc = __builtin_amdgcn_wmma_f32_16x16x4_f32(
            false, a, false, b, (short)0, acc, false, false);
    }

#pragma unroll
    for (int v = 0; v < 8; ++v) {
        t2[(size_t)(m0 + v + half * 8) * NCLASS + n] = acc[v];
    }
}

// ---------------------------------------------------------------------------
// COO SpMM: out[row[e], :] += vals[e] * dense[col[e], :]
// One thread per (edge, float4 chunk). LOGQ = log2(D/4).
// Gathers + atomics are L2-resident (dense matrix fits in 192MB L2).
// ---------------------------------------------------------------------------
template <int LOGQ, int D>
__global__ void __launch_bounds__(256)
gcn_spmm_atomic(const int* __restrict__ row,
                const int* __restrict__ col,
                const float* __restrict__ vals,
                const float* __restrict__ dense,
                float* __restrict__ out,
                long long total) {
    const long long idx = (long long)blockIdx.x * blockDim.x + threadIdx.x;
    if (idx >= total) return;
    const int e  = (int)(idx >> LOGQ);
    const int f4 = ((int)idx & ((1 << LOGQ) - 1)) << 2;
    const int r = row[e];
    const int c = col[e];
    const float v = vals[e];
    const float4 d = *(const float4*)(dense + (size_t)c * D + f4);
    float* o = out + (size_t)r * D + f4;
    unsafeAtomicAdd(o + 0, v * d.x);
    unsafeAtomicAdd(o + 1, v * d.y);
    unsafeAtomicAdd(o + 2, v * d.z);
    unsafeAtomicAdd(o + 3, v * d.w);
}

// ---------------------------------------------------------------------------
// Launch
// ---------------------------------------------------------------------------
extern "C" void kernel_launch(void* const* d_in, const int* in_sizes, int n_in,
                              void* d_out, int out_size, void* d_ws, size_t ws_size,
                              hipStream_t stream) {
    const float* x      = (const float*)d_in[0];
    const int*   row    = (const int*)d_in[1];
    const int*   col    = (const int*)d_in[2];
    const float* vals   = (const float*)d_in[3];
    const float* mask1  = (const float*)d_in[4];
    const float* mask2  = (const float*)d_in[5];
    const float* W1     = (const float*)d_in[6];
    const float* W2     = (const float*)d_in[7];
    float* out = (float*)d_out;

    float* t1 = (float*)d_ws;                                   // [N, NHID]
    float* h1 = t1 + (size_t)NNODES * NHID;                     // [N, NHID]
    float* t2 = h1 + (size_t)NNODES * NHID;                     // [N, NCLASS]

    const int mblocks = NNODES / 16;                            // 3125 (exact)

    // Zero accumulation targets (graph-capturable async memset)
    hipMemsetAsync(h1, 0, (size_t)NNODES * NHID * sizeof(float), stream);
    hipMemsetAsync(out, 0, (size_t)NNODES * NCLASS * sizeof(float), stream);

    // t1 = dropout(x) @ W1
    gcn_gemm1_dropout_wmma<<<mblocks, 128, 0, stream>>>(x, mask1, W1, t1);

    // h1 = A @ t1
    {
        const long long total = (long long)NEDGES * (NHID / 4);  // 51.2M
        const int blocks = (int)((total + 255) / 256);
        gcn_spmm_atomic<6, NHID><<<blocks, 256, 0, stream>>>(row, col, vals, t1, h1, total);
    }

    // t2 = dropout(relu(h1)) @ W2
    gcn_gemm2_relu_dropout_wmma<<<mblocks, 128, 0, stream>>>(h1, mask2, W2, t2);

    // out = A @ t2
    {
        const long long total = (long long)NEDGES * (NCLASS / 4); // 12.8M
        const int blocks = (int)((total + 255) / 256);
        gcn_spmm_atomic<4, NCLASS><<<blocks, 256, 0, stream>>>(row, col, vals, t2, out, total);
    }
}